// Method_GCN_Class_74414603370819
// MI455X (gfx1250) — compile-verified
//
#include <hip/hip_runtime.h>
#include <hip/hip_bf16.h>
#include <math.h>

// GCN 2-layer forward for MI455X (gfx1250, wave32).
// Pipeline per call (all on `stream`, graph-capture safe):
//   1) counts=0 ; histogram(edge_row) ; 3-kernel exclusive scan -> row_ptr, cursor
//   2) scatter edges into CSR order (colS, valS) via integer atomics
//   3) support1 = x @ w1                  (fp32 WMMA 16x16x4, NN/KK compile-time)
//   4) h = relu(CSR-SpMM(support1) + b1)  (LDS-staged edge lists, no fp atomics;
//                                          support1 = 102MB -> L2-resident)
//   5) w2p = zero-pad(w2) to 256x48 ; support2p = h @ w2p  (stride 48, no predication)
//   6) d_out = CSR-SpMM(support2p) + b2   (one wave per row, lanes cover 40 feats)
//   7) log_softmax rows of 40 in place
// Workspace use: ~250 MB.

typedef __attribute__((ext_vector_type(2))) float v2f;
typedef __attribute__((ext_vector_type(8))) float v8f;

#define GCN_IN_FEAT 512
#define GCN_HIDDEN  256
#define GCN_NCLASS  40
#define GCN_NPAD    48
#define SCAN_CHUNK  1024

// ---------------------------------------------------------------- histogram
__global__ __launch_bounds__(256) void k_hist(const int* __restrict__ rows,
                                              int* __restrict__ counts, int E) {
  int i = blockIdx.x * 256 + threadIdx.x;
  if (i < E) atomicAdd(&counts[rows[i]], 1);
}

// ------------------------------------------------------------- scan stage 1
__global__ __launch_bounds__(256) void k_scan_partial(const int* __restrict__ counts,
                                                      int* __restrict__ partials, int N) {
  __shared__ int sh[256];
  int t = threadIdx.x;
  int base = blockIdx.x * SCAN_CHUNK;
  int s = 0;
#pragma unroll
  for (int j = 0; j < 4; ++j) {
    int i = base + j * 256 + t;
    s += (i < N) ? counts[i] : 0;
  }
  sh[t] = s;
  __syncthreads();
  for (int off = 128; off > 0; off >>= 1) {
    if (t < off) sh[t] += sh[t + off];
    __syncthreads();
  }
  if (t == 0) partials[blockIdx.x] = sh[0];
}

// ------------------------------------------------------------- scan stage 2
// nB <= 128 for N <= 131072.
__global__ __launch_bounds__(128) void k_scan_small(int* __restrict__ partials,
                                                    int* __restrict__ row_ptr,
                                                    int nB, int N) {
  __shared__ int sh[128];
  int t = threadIdx.x;
  int v = (t < nB) ? partials[t] : 0;
  sh[t] = v;
  __syncthreads();
  for (int off = 1; off < 128; off <<= 1) {
    int u = (t >= off) ? sh[t - off] : 0;
    __syncthreads();
    sh[t] += u;
    __syncthreads();
  }
  if (t < nB) partials[t] = sh[t] - v;   // exclusive
  if (t == 127) row_ptr[N] = sh[127];    // total edge count
}

// ------------------------------------------------------------- scan stage 3
__global__ __launch_bounds__(256) void k_scan_final(const int* __restrict__ counts,
                                                    const int* __restrict__ partials,
                                                    int* __restrict__ row_ptr,
                                                    int* __restrict__ cursor, int N) {
  __shared__ int sh[256];
  int t = threadIdx.x;
  int base = blockIdx.x * SCAN_CHUNK + t * 4;
  int l[4];
  int s = 0;
#pragma unroll
  for (int j = 0; j < 4; ++j) {
    int i = base + j;
    l[j] = (i < N) ? counts[i] : 0;
    s += l[j];
  }
  sh[t] = s;
  __syncthreads();
  for (int off = 1; off < 256; off <<= 1) {
    int u = (t >= off) ? sh[t - off] : 0;
    __syncthreads();
    sh[t] += u;
    __syncthreads();
  }
  int run = partials[blockIdx.x] + (sh[t] - s);
#pragma unroll
  for (int j = 0; j < 4; ++j) {
    int i = base + j;
    if (i < N) { row_ptr[i] = run; cursor[i] = run; }
    run += l[j];
  }
}

// ---------------------------------------------------------------- scatter
__global__ __launch_bounds__(256) void k_scatter(const int* __restrict__ rows,
                                                 const int* __restrict__ cols,
                                                 const float* __restrict__ vals,
                                                 int* __restrict__ cursor,
                                                 int* __restrict__ colS,
                                                 float* __restrict__ valS, int E) {
  int i = blockIdx.x * 256 + threadIdx.x;
  if (i < E) {
    int r = rows[i];
    int p = atomicAdd(&cursor[r], 1);
    colS[p] = cols[i];
    valS[p] = vals[i];
  }
}

// ------------------------------------------------------------- pad w2 -> 256x48
__global__ __launch_bounds__(256) void k_pad_w2(const float* __restrict__ w2,
                                                float* __restrict__ w2p) {
  int i = blockIdx.x * 256 + threadIdx.x;          // over HIDDEN*NPAD = 12288
  if (i < GCN_HIDDEN * GCN_NPAD) {
    int r = i / GCN_NPAD, c = i % GCN_NPAD;
    w2p[i] = (c < GCN_NCLASS) ? w2[r * GCN_NCLASS + c] : 0.f;
  }
}

// --------------------------------------------------- dense GEMM, fp32 WMMA
// C[M x NN] = A[M x KK] @ B[KK x NN], all strides compile-time.
// One 16x16 C tile per wave; branch-free full-EXEC inner loop.
// A frag (16x4 f32): lane mn=lane&15 -> row, kh=lane>>4; a[i]=A[row][k0+2*kh+i]
// B frag (4x16 f32): b[i]=B[k0+2*kh+i][col], col = tN*16 + mn
// C/D (16x16 f32, 8 VGPRs): c[j] -> row tM*16 + 8*kh + j, col as above
template <int NN, int KK>
__global__ __launch_bounds__(256) void k_gemm_wmma_f32(const float* __restrict__ A,
                                                       const float* __restrict__ B,
                                                       float* __restrict__ C,
                                                       int tilesN, int totalWaves) {
  int wid = blockIdx.x * (blockDim.x >> 5) + (threadIdx.x >> 5);
  if (wid >= totalWaves) return;                 // wave-uniform exit
  int lane = threadIdx.x & 31;
  int tN = wid % tilesN;
  int tM = wid / tilesN;
  int mn = lane & 15;
  int kh = lane >> 4;
  int col = tN * 16 + mn;

  const float* aP = A + (size_t)(tM * 16 + mn) * KK + 2 * kh;
  const float* bP = B + (size_t)(2 * kh) * NN + col;

  v8f acc = {0.f, 0.f, 0.f, 0.f, 0.f, 0.f, 0.f, 0.f};
#pragma unroll 8
  for (int k0 = 0; k0 < KK; k0 += 4) {
    v2f a = {aP[0], aP[1]};
    v2f b = {bP[0], bP[NN]};
    acc = __builtin_amdgcn_wmma_f32_16x16x4_f32(false, a, false, b,
                                                (short)0, acc, false, false);
    aP += 4;
    bP += 4 * NN;
  }
  float* cP = C + (size_t)(tM * 16 + 8 * kh) * NN + col;
#pragma unroll
  for (int j = 0; j < 8; ++j) cP[(size_t)j * NN] = acc[j];
}

// --------------------------------------------- CSR SpMM, 256 features/row
// One block per row, one feature per thread; edge (col,val) chunks staged in
// LDS so per-edge broadcasts come from ds_read, not duplicated vmem requests.
// Trip counts are block-uniform (same row) -> barriers are safe.
__global__ __launch_bounds__(256) void k_spmm256_bias_relu(const int* __restrict__ rp,
                                                           const int* __restrict__ colS,
                                                           const float* __restrict__ valS,
                                                           const float* __restrict__ S,
                                                           const float* __restrict__ bias,
                                                           float* __restrict__ H) {
  __shared__ int   shc[256];
  __shared__ float shv[256];
  int r = blockIdx.x;
  int f = threadIdx.x;
  int s = rp[r], e = rp[r + 1];
  float acc = 0.f;
  for (int base = s; base < e; base += 256) {
    int n = min(256, e - base);
    if (threadIdx.x < n) {
      shc[threadIdx.x] = colS[base + threadIdx.x];
      shv[threadIdx.x] = valS[base + threadIdx.x];
    }
    __syncthreads();
    for (int i = 0; i < n; ++i) {
      acc = fmaf(shv[i], S[(size_t)shc[i] * GCN_HIDDEN + f], acc);
    }
    __syncthreads();
  }
  float o = acc + bias[f];
  H[(size_t)r * GCN_HIDDEN + f] = fmaxf(o, 0.f);
}

// ---------------------------------------------- CSR SpMM, 40 features/row
// One wave per row. Lane l accumulates features l and l+32 (lanes >= 8 read
// in-bounds padding garbage for the second slot; result discarded).
// S has row stride GCN_NPAD=48 and the allocation is over-padded by 64 floats.
__global__ __launch_bounds__(256) void k_spmm40_bias(const int* __restrict__ rp,
                                                     const int* __restrict__ colS,
                                                     const float* __restrict__ valS,
                                                     const float* __restrict__ S,
                                                     const float* __restrict__ bias,
                                                     float* __restrict__ Out, int N) {
  int r = blockIdx.x * 8 + (threadIdx.x >> 5);
  if (r >= N) return;                            // wave-uniform
  int lane = threadIdx.x & 31;
  int s = rp[r], e = rp[r + 1];
  float acc1 = 0.f, acc2 = 0.f;
  for (int i = s; i < e; ++i) {
    int   c = colS[i];                           // wave-broadcast load
    float v = valS[i];
    const float* row = S + (size_t)c * GCN_NPAD;
    acc1 = fmaf(v, row[lane], acc1);
    acc2 = fmaf(v, row[32 + lane], acc2);
  }
  Out[(size_t)r * GCN_NCLASS + lane] = acc1 + bias[lane];
  if (lane < GCN_NCLASS - 32) {
    Out[(size_t)r * GCN_NCLASS + 32 + lane] = acc2 + bias[32 + lane];
  }
}

// -------------------------------------------------- row log_softmax (40)
__global__ __launch_bounds__(128) void k_log_softmax40(float* __restrict__ Out, int N) {
  int r = blockIdx.x * 128 + threadIdx.x;
  if (r >= N) return;
  float* p = Out + (size_t)r * GCN_NCLASS;
  float v[GCN_NCLASS];
  float mx = -INFINITY;
#pragma unroll
  for (int i = 0; i < GCN_NCLASS; ++i) {
    v[i] = p[i];
    mx = fmaxf(mx, v[i]);
  }
  float se = 0.f;
#pragma unroll
  for (int i = 0; i < GCN_NCLASS; ++i) se += expf(v[i] - mx);
  float lse = mx + logf(se);
#pragma unroll
  for (int i = 0; i < GCN_NCLASS; ++i) p[i] = v[i] - lse;
}

extern "C" void kernel_launch(void* const* d_in, const int* in_sizes, int n_in,
                              void* d_out, int out_size, void* d_ws, size_t ws_size,
                              hipStream_t stream) {
  const float* x  = (const float*)d_in[0];
  const float* w1 = (const float*)d_in[1];
  const float* b1 = (const float*)d_in[2];
  const float* w2 = (const float*)d_in[3];
  const float* b2 = (const float*)d_in[4];
  const float* ev = (const float*)d_in[5];
  const int*   er = (const int*)d_in[6];
  const int*   ec = (const int*)d_in[7];

  const int N = in_sizes[0] / GCN_IN_FEAT;   // 100000 (divisible by 16)
  const int E = in_sizes[5];                 // 3200000
  float* out = (float*)d_out;

  // ---- workspace carve-out (256B aligned), total ~250 MB ----
  char* ws = (char*)d_ws;
  size_t off = 0;
  auto carve = [&](size_t bytes) -> char* {
    char* p = ws + off;
    off = (off + bytes + 255) & ~(size_t)255;
    return p;
  };
  float* support1  = (float*)carve((size_t)N * GCN_HIDDEN * 4);
  float* h         = (float*)carve((size_t)N * GCN_HIDDEN * 4);
  float* support2p = (float*)carve(((size_t)N * GCN_NPAD + 64) * 4);
  int*   colS      = (int*)carve((size_t)E * 4);
  float* valS      = (float*)carve((size_t)E * 4);
  int*   row_ptr   = (int*)carve((size_t)(N + 1) * 4);
  int*   counts    = (int*)carve((size_t)N * 4);
  int*   cursor    = (int*)carve((size_t)N * 4);
  int*   partials  = (int*)carve(128 * 4);
  float* w2p       = (float*)carve((size_t)GCN_HIDDEN * GCN_NPAD * 4);
  (void)ws_size; (void)n_in; (void)out_size;

  // ---- CSR build ----
  hipMemsetAsync(counts, 0, (size_t)N * 4, stream);
  k_hist<<<(E + 255) / 256, 256, 0, stream>>>(er, counts, E);
  int nB = (N + SCAN_CHUNK - 1) / SCAN_CHUNK;  // 98 <= 128
  k_scan_partial<<<nB, 256, 0, stream>>>(counts, partials, N);
  k_scan_small<<<1, 128, 0, stream>>>(partials, row_ptr, nB, N);
  k_scan_final<<<nB, 256, 0, stream>>>(counts, partials, row_ptr, cursor, N);
  k_scatter<<<(E + 255) / 256, 256, 0, stream>>>(er, ec, ev, cursor, colS, valS, E);

  // ---- layer 1: dense transform + aggregate + bias + relu ----
  {
    int tilesN = GCN_HIDDEN / 16;            // 16
    int totalWaves = (N / 16) * tilesN;      // 100000
    int blocks = (totalWaves + 7) / 8;       // 8 waves per 256-thread block
    k_gemm_wmma_f32<GCN_HIDDEN, GCN_IN_FEAT>
        <<<blocks, 256, 0, stream>>>(x, w1, support1, tilesN, totalWaves);
  }
  k_spmm256_bias_relu<<<N, 256, 0, stream>>>(row_ptr, colS, valS, support1, b1, h);

  // ---- layer 2: pad w2, dense transform (stride 48), aggregate + bias ----
  k_pad_w2<<<(GCN_HIDDEN * GCN_NPAD + 255) / 256, 256, 0, stream>>>(w2, w2p);
  {
    int tilesN = GCN_NPAD / 16;              // 3
    int totalWaves = (N / 16) * tilesN;      // 18750
    int blocks = (totalWaves + 7) / 8;
    k_gemm_wmma_f32<GCN_NPAD, GCN_HIDDEN>
        <<<blocks, 256, 0, stream>>>(h, w2p, support2p, tilesN, totalWaves);
  }
  k_spmm40_bias<<<(N + 7) / 8, 256, 0, stream>>>(row_ptr, colS, valS, support2p, b2,
                                                 out, N);

  // ---- log_softmax ----
  k_log_softmax40<<<(N + 127) / 128, 128, 0, stream>>>(out, N);
}